// HyenaLayer_601295421644
// MI455X (gfx1250) — compile-verified
//
#include <hip/hip_runtime.h>
#include <math.h>
#include <stdint.h>

typedef __attribute__((ext_vector_type(2))) float v2f;
typedef __attribute__((ext_vector_type(8))) float v8f;
typedef __attribute__((address_space(3))) float lds_f;

#define N_STAGES 3
#define DCH      1024
#define TSEQ     4096
#define BATCH    4
#define LSIZE    8192          // L = 2*T (causal zero-pad)
#define LOG2L    13
#define PI_F     3.14159265358979323846f

// GEMM block tiling
#define BM 128
#define BN 64
#define KS 16
#define APAD 20                // 128x16 A tile rows padded to 20 floats (16B aligned, bank-conflict-free)
#define BPAD 80                // 16x64  B tile rows padded to 80 floats

// ---------------------------------------------------------------------------
// LDS-tiled fp32 WMMA GEMM: C = A[MxK] * B[KxN] + bias[N].
// 256 threads = 8 waves in a 4x2 grid; each wave computes a 32x32 register
// tile (2x2 WMMA accumulators) of the block's 128x64 tile. K stepped by 16.
// colMajor=1 stores C transposed (C[n*M+m]) for the downstream FFT kernels.
// ---------------------------------------------------------------------------
__global__ void wmma_gemm_bias(const float* __restrict__ A,
                               const float* __restrict__ Bm,
                               const float* __restrict__ bias,
                               float* __restrict__ C,
                               int M, int N, int K, int colMajor) {
  __shared__ float As[BM][APAD];
  __shared__ float Bs[KS][BPAD];
  const int tid  = threadIdx.x;
  const int lane = tid & 31;
  const int wave = tid >> 5;
  const int waveM = wave & 3;            // 0..3 -> 32-row slice
  const int waveN = wave >> 2;           // 0..1 -> 32-col slice
  const int l15 = lane & 15;
  const int hi  = lane >> 4;             // 0: K={0,1}, 1: K={2,3}
  const int bm0 = blockIdx.y * BM;
  const int bn0 = blockIdx.x * BN;

  const int arow = (tid * 2) >> 2;       // A-load: rows for this thread's 2 float4s
  const int brow = tid >> 4;             // B-load row
  const int bseg = tid & 15;

  v8f acc[2][2] = {};
  for (int k0 = 0; k0 < K; k0 += KS) {
    // stage A tile (128x16): 2 float4 per thread
#pragma unroll
    for (int i = 0; i < 2; ++i) {
      const int li  = tid * 2 + i;
      const int row = li >> 2;
      const int seg = li & 3;
      const float4 v = *(const float4*)(A + (size_t)(bm0 + row) * K + k0 + seg * 4);
      *(float4*)&As[row][seg * 4] = v;
    }
    // stage B tile (16x64): 1 float4 per thread
    {
      const float4 v = *(const float4*)(Bm + (size_t)(k0 + brow) * N + bn0 + bseg * 4);
      *(float4*)&Bs[brow][bseg * 4] = v;
    }
    __builtin_prefetch((const void*)(A + (size_t)(bm0 + arow) * K + k0 + KS), 0, 0);
    __builtin_prefetch((const void*)(Bm + (size_t)(k0 + KS + brow) * N + bn0), 0, 0);
    __syncthreads();

#pragma unroll
    for (int k = 0; k < KS; k += 4) {
      v2f af[2], bfr[2];
#pragma unroll
      for (int r = 0; r < 2; ++r) {
        const int row = waveM * 32 + r * 16 + l15;
        af[r] = *(const v2f*)&As[row][k + 2 * hi];
      }
#pragma unroll
      for (int c = 0; c < 2; ++c) {
        const int col = waveN * 32 + c * 16 + l15;
        bfr[c].x = Bs[k + 2 * hi][col];
        bfr[c].y = Bs[k + 2 * hi + 1][col];
      }
#pragma unroll
      for (int r = 0; r < 2; ++r)
#pragma unroll
        for (int c = 0; c < 2; ++c)
          acc[r][c] = __builtin_amdgcn_wmma_f32_16x16x4_f32(
              false, af[r], false, bfr[c], (short)0, acc[r][c], false, false);
    }
    __syncthreads();
  }

#pragma unroll
  for (int r = 0; r < 2; ++r)
#pragma unroll
    for (int c = 0; c < 2; ++c) {
      const int col = bn0 + waveN * 32 + c * 16 + l15;
      const float bv = bias[col];
#pragma unroll
      for (int rr = 0; rr < 8; ++rr) {
        const int row = bm0 + waveM * 32 + r * 16 + rr + (hi << 3);
        const size_t idx = colMajor ? ((size_t)col * M + row)
                                    : ((size_t)row * N + col);
        C[idx] = acc[r][c][rr] + bv;
      }
    }
}

// ---------------------------------------------------------------------------
// Same tiled GEMM but A = sinusoidal positional encoding generated in-LDS.
// pos[t,k] = sin(t*10000^{-k/D}) (k even) / cos(t*10000^{-(k-1)/D}) (k odd)
// Output stored transposed: F_t[(s*D+d)*T + t].
// ---------------------------------------------------------------------------
__global__ void wmma_gemm_pos_bias(const float* __restrict__ Wf,
                                   const float* __restrict__ bf,
                                   float* __restrict__ C,
                                   int M, int N, int K) {
  __shared__ float As[BM][APAD];
  __shared__ float Bs[KS][BPAD];
  const int tid  = threadIdx.x;
  const int lane = tid & 31;
  const int wave = tid >> 5;
  const int waveM = wave & 3;
  const int waveN = wave >> 2;
  const int l15 = lane & 15;
  const int hi  = lane >> 4;
  const int bm0 = blockIdx.y * BM;
  const int bn0 = blockIdx.x * BN;
  const float c0 = -9.210340371976184f / (float)K;   // -ln(10000)/D

  const int brow = tid >> 4;
  const int bseg = tid & 15;

  v8f acc[2][2] = {};
  for (int k0 = 0; k0 < K; k0 += KS) {
    // generate A tile (pos encoding): 2 float4 per thread
#pragma unroll
    for (int i = 0; i < 2; ++i) {
      const int li  = tid * 2 + i;
      const int row = li >> 2;
      const int seg = li & 3;
      const int kk  = k0 + seg * 4;                  // even
      const float t = (float)(bm0 + row);
      const float a0 = t * __expf(c0 * (float)kk);
      const float a1 = t * __expf(c0 * (float)(kk + 2));
      float s0, cS0, s1, cS1;
      __sincosf(a0, &s0, &cS0);
      __sincosf(a1, &s1, &cS1);
      float4 v; v.x = s0; v.y = cS0; v.z = s1; v.w = cS1;
      *(float4*)&As[row][seg * 4] = v;
    }
    {
      const float4 v = *(const float4*)(Wf + (size_t)(k0 + brow) * N + bn0 + bseg * 4);
      *(float4*)&Bs[brow][bseg * 4] = v;
    }
    __syncthreads();

#pragma unroll
    for (int k = 0; k < KS; k += 4) {
      v2f af[2], bfr[2];
#pragma unroll
      for (int r = 0; r < 2; ++r) {
        const int row = waveM * 32 + r * 16 + l15;
        af[r] = *(const v2f*)&As[row][k + 2 * hi];
      }
#pragma unroll
      for (int c = 0; c < 2; ++c) {
        const int col = waveN * 32 + c * 16 + l15;
        bfr[c].x = Bs[k + 2 * hi][col];
        bfr[c].y = Bs[k + 2 * hi + 1][col];
      }
#pragma unroll
      for (int r = 0; r < 2; ++r)
#pragma unroll
        for (int c = 0; c < 2; ++c)
          acc[r][c] = __builtin_amdgcn_wmma_f32_16x16x4_f32(
              false, af[r], false, bfr[c], (short)0, acc[r][c], false, false);
    }
    __syncthreads();
  }

#pragma unroll
  for (int r = 0; r < 2; ++r)
#pragma unroll
    for (int c = 0; c < 2; ++c) {
      const int col = bn0 + waveN * 32 + c * 16 + l15;
      const float bv = bf[col];
#pragma unroll
      for (int rr = 0; rr < 8; ++rr) {
        const int row = bm0 + waveM * 32 + r * 16 + rr + (hi << 3);
        C[(size_t)col * M + row] = acc[r][c][rr] + bv;   // transposed store
      }
    }
}

// ---------------------------------------------------------------------------
// Radix-2 FFT with LDS twiddle table tw[j] = exp(-2*pi*i*j/L), j < L/2.
// fft_dif : natural in  -> bit-reversed out, forward.
// fft_dit : bit-reversed in -> natural out, inverse (conjugate), unscaled.
// Spectral multiply happens in bit-reversed domain -> no reorder pass.
// ---------------------------------------------------------------------------
__device__ __forceinline__ void twiddle_init(float2* tw) {
  for (int j = threadIdx.x; j < (LSIZE >> 1); j += 256) {
    float sn, cs;
    __sincosf(-2.0f * PI_F * (float)j / (float)LSIZE, &sn, &cs);
    tw[j] = make_float2(cs, sn);
  }
}

__device__ __forceinline__ void fft_dif(float2* s, const float2* tw) {
  for (int lh = LOG2L - 1; lh >= 0; --lh) {
    const int half = 1 << lh;
    for (int j = threadIdx.x; j < (LSIZE >> 1); j += 256) {
      const int k   = j & (half - 1);
      const int idx = ((j >> lh) << (lh + 1)) + k;
      const float2 u = s[idx];
      const float2 v = s[idx + half];
      const float2 w = tw[k << (LOG2L - 1 - lh)];
      const float dx = u.x - v.x, dy = u.y - v.y;
      s[idx]        = make_float2(u.x + v.x, u.y + v.y);
      s[idx + half] = make_float2(dx * w.x - dy * w.y, dx * w.y + dy * w.x);
    }
    __syncthreads();
  }
}

__device__ __forceinline__ void fft_dit_inv(float2* s, const float2* tw) {
  for (int lh = 0; lh < LOG2L; ++lh) {
    const int half = 1 << lh;
    for (int j = threadIdx.x; j < (LSIZE >> 1); j += 256) {
      const int k   = j & (half - 1);
      const int idx = ((j >> lh) << (lh + 1)) + k;
      const float2 w = tw[k << (LOG2L - 1 - lh)];       // conj applied below
      const float2 v = s[idx + half];
      const float2 t = make_float2(v.x * w.x + v.y * w.y,
                                   v.y * w.x - v.x * w.y);
      const float2 u = s[idx];
      s[idx]        = make_float2(u.x + t.x, u.y + t.y);
      s[idx + half] = make_float2(u.x - t.x, u.y - t.y);
    }
    __syncthreads();
  }
}

// ---------------------------------------------------------------------------
// Async global->LDS copy of one contiguous 16KB row (4096 floats) using
// GLOBAL_LOAD_ASYNC_TO_LDS_B128 (ASYNCcnt-tracked). Each of the 256 threads
// issues 4 lane-contiguous b128 transfers. Drain with s_wait_asynccnt 0.
// ---------------------------------------------------------------------------
__device__ __forceinline__ void async_copy_row16k(const float* __restrict__ g,
                                                  float* lds_dst) {
  const int tid = threadIdx.x;
#pragma unroll
  for (int j = 0; j < 4; ++j) {
    const int idx4 = j * 256 + tid;                  // float4 index in the row
    const uint32_t loff =
        (uint32_t)(uintptr_t)(lds_f*)(lds_dst + (size_t)idx4 * 4);
    const uint64_t gaddr = (uint64_t)(uintptr_t)(g + (size_t)idx4 * 4);
    asm volatile("global_load_async_to_lds_b128 %0, %1, off"
                 :: "v"(loff), "v"(gaddr) : "memory");
  }
}

__device__ __forceinline__ void async_drain() {
  asm volatile("s_wait_asynccnt 0x0" ::: "memory");
}

// ---------------------------------------------------------------------------
// Filter spectra: F_br[s*D+d][k] = DIF-FFT of zero-padded f[:, s, d],
// kept in bit-reversed order. One workgroup per (stage, channel).
// ---------------------------------------------------------------------------
__global__ void filter_fft(const float* __restrict__ fm_t,
                           float* __restrict__ Fbr) {
  __shared__ float2 s[LSIZE];
  __shared__ float2 tw[LSIZE >> 1];
  const int sd = blockIdx.x;                       // 0 .. 3*D-1
  twiddle_init(tw);
  const float* __restrict__ src = fm_t + (size_t)sd * TSEQ;
  for (int t = threadIdx.x; t < LSIZE; t += 256)
    s[t] = make_float2(t < TSEQ ? src[t] : 0.0f, 0.0f);
  __syncthreads();
  fft_dif(s, tw);
  float2* __restrict__ dst = (float2*)Fbr + (size_t)sd * LSIZE;
  for (int t = threadIdx.x; t < LSIZE; t += 256)
    dst[t] = s[t];
}

// ---------------------------------------------------------------------------
// Fused 3-stage Hyena FFT-conv + gating. One workgroup per (b, d) sequence.
// LDS: 64KB data + 32KB twiddles + 16KB async gate staging = 112KB
// (2 workgroups per 320KB WGP). Gate rows for stage i are prefetched with
// async b128 copies while stage i's FFTs run; drained via s_wait_asynccnt.
// xp_t layout: [s*D+d][b*T + t]  (transposed store from GEMM 1).
// ---------------------------------------------------------------------------
__global__ void hyena_conv(const float* __restrict__ xp_t,
                           const float* __restrict__ Fbr,
                           float* __restrict__ out) {
  __shared__ float2 s[LSIZE];
  __shared__ float2 tw[LSIZE >> 1];
  __shared__ float  gbuf[TSEQ];
  const int b = blockIdx.x / DCH;
  const int d = blockIdx.x % DCH;
  const size_t Mrows = (size_t)BATCH * TSEQ;
  const float invL = 1.0f / (float)LSIZE;

  twiddle_init(tw);
  const float* __restrict__ x0 = xp_t + (size_t)d * Mrows + (size_t)b * TSEQ;
  for (int t = threadIdx.x; t < LSIZE; t += 256)
    s[t] = make_float2(t < TSEQ ? x0[t] : 0.0f, 0.0f);
  // prefetch stage-0 gate row while the first forward FFT runs
  async_copy_row16k(xp_t + (size_t)(1 * DCH + d) * Mrows + (size_t)b * TSEQ, gbuf);
  __syncthreads();

  for (int st = 0; st < N_STAGES; ++st) {
    fft_dif(s, tw);
    const float2* __restrict__ Fk =
        (const float2*)Fbr + (size_t)(st * DCH + d) * LSIZE;
    for (int t = threadIdx.x; t < LSIZE; t += 256) {
      const float2 a = s[t];
      const float2 w = Fk[t];
      s[t] = make_float2(a.x * w.x - a.y * w.y, a.x * w.y + a.y * w.x);
    }
    __syncthreads();
    fft_dit_inv(s, tw);

    async_drain();                 // this wave's gate fills have landed
    __syncthreads();               // all waves' fills visible to everyone
    for (int t = threadIdx.x; t < LSIZE; t += 256) {
      const float gv = (t < TSEQ) ? gbuf[t] : 0.0f;
      s[t] = make_float2(s[t].x * invL * gv, 0.0f);
    }
    __syncthreads();               // gbuf fully consumed by all waves
    if (st + 1 < N_STAGES)         // refill for next stage; overlaps next FFT
      async_copy_row16k(xp_t + (size_t)((st + 2) * DCH + d) * Mrows
                             + (size_t)b * TSEQ, gbuf);
  }

  for (int t = threadIdx.x; t < TSEQ; t += 256)
    out[((size_t)b * TSEQ + t) * DCH + d] = s[t].x;
}

// ---------------------------------------------------------------------------
// Launch
// ---------------------------------------------------------------------------
extern "C" void kernel_launch(void* const* d_in, const int* in_sizes, int n_in,
                              void* d_out, int out_size, void* d_ws, size_t ws_size,
                              hipStream_t stream) {
  const float* X  = (const float*)d_in[0];   // (4, 4096, 1024)
  const float* Wp = (const float*)d_in[1];   // (1024, 4096)
  const float* bp = (const float*)d_in[2];   // (4096,)
  const float* Wf = (const float*)d_in[3];   // (1024, 3072)
  const float* bf = (const float*)d_in[4];   // (3072,)
  float* out = (float*)d_out;

  const int M1 = BATCH * TSEQ;               // 16384
  const int N1 = (N_STAGES + 1) * DCH;       // 4096
  const int M2 = TSEQ;                       // 4096
  const int N2 = N_STAGES * DCH;             // 3072
  const int K  = DCH;                        // 1024

  float* ws   = (float*)d_ws;
  float* xp_t = ws;                                   // N1 x M1  (transposed xp)
  float* fm_t = xp_t + (size_t)N1 * M1;               // N2 x M2  (transposed f)
  float* Fbr  = fm_t + (size_t)N2 * M2;               // N2 x L complex (bit-rev)

  // xp^T = (X @ Wp + bp)^T : grid 64 x 128 blocks of 128x64
  wmma_gemm_bias<<<dim3(N1 / BN, M1 / BM), dim3(256), 0, stream>>>(
      X, Wp, bp, xp_t, M1, N1, K, /*colMajor=*/1);
  // f^T = (pos @ Wf + bf)^T : grid 48 x 32
  wmma_gemm_pos_bias<<<dim3(N2 / BN, M2 / BM), dim3(256), 0, stream>>>(
      Wf, bf, fm_t, M2, N2, K);
  // filter spectra (bit-reversed order)
  filter_fft<<<dim3(N_STAGES * DCH), dim3(256), 0, stream>>>(fm_t, Fbr);
  // fused 3-stage FFT-conv + gate
  hyena_conv<<<dim3(BATCH * DCH), dim3(256), 0, stream>>>(xp_t, Fbr, out);
}